// SpecAugment_75239237092009
// MI455X (gfx1250) — compile-verified
//
#include <hip/hip_runtime.h>
#include <cstdint>
#include <cstddef>

typedef __attribute__((ext_vector_type(4))) float v4f;
typedef __attribute__((ext_vector_type(4))) int   v4i;

#define B_DIM   128
#define T_DIM   4000
#define F_DIM   80
#define ROW4    (F_DIM / 4)        // 20 float4 per (b,t) row
#define N4      (T_DIM * ROW4)     // 80000 float4 per batch element
#define THREADS 256
#define ITEMS   4
#define TILE    (THREADS * ITEMS)  // 1024 float4 per block

// CDNA5 async global->LDS path (ASYNCcnt), guarded so the file compiles on
// host pass / toolchains without the builtin.
#if defined(__gfx1250__) &&                                                  \
    __has_builtin(__builtin_amdgcn_global_load_async_to_lds_b128) &&         \
    __has_builtin(__builtin_amdgcn_s_wait_asynccnt)
#define USE_ASYNC 1
// Exact parameter types per hipcc diagnostic: (v4i AS1*, v4i AS3*, imm, imm)
typedef __attribute__((address_space(1))) v4i* gbl_v4i_p;
typedef __attribute__((address_space(3))) v4i* lds_v4i_p;
#else
#define USE_ASYNC 0
#endif

__global__ __launch_bounds__(THREADS) void SpecAugment_75239237092009_kernel(
    const float* __restrict__ xin,
    const int* __restrict__ f_width,
    const int* __restrict__ f_start,
    const int* __restrict__ t_width,
    const int* __restrict__ t_start,
    float* __restrict__ outp)
{
    const int b = blockIdx.y;  // uniform -> scalar loads of mask params

    const int fs0 = f_start[2 * b + 0], fe0 = fs0 + f_width[2 * b + 0];
    const int fs1 = f_start[2 * b + 1], fe1 = fs1 + f_width[2 * b + 1];
    const int ts0 = t_start[2 * b + 0], te0 = ts0 + t_width[2 * b + 0];
    const int ts1 = t_start[2 * b + 1], te1 = ts1 + t_width[2 * b + 1];

    // Per-batch frequency keep mask, built once per block, read as b128.
    __shared__ __align__(16) float kf[F_DIM];
    if (threadIdx.x < F_DIM) {
        const int f = threadIdx.x;
        const bool fm = (f >= fs0 && f < fe0) || (f >= fs1 && f < fe1);
        kf[f] = fm ? 0.0f : 1.0f;
    }
#if USE_ASYNC
    __shared__ __align__(16) v4f stage[TILE];  // 16 KB staging buffer
#endif
    __syncthreads();

    const v4f* __restrict__ xb = (const v4f*)xin + (size_t)b * N4;
    v4f* __restrict__ ob       = (v4f*)outp      + (size_t)b * N4;
    const int base = blockIdx.x * TILE + (int)threadIdx.x;

#if USE_ASYNC
    // Issue all staging loads up front (tracked by ASYNCcnt, in-order).
#pragma unroll
    for (int i = 0; i < ITEMS; ++i) {
        const int i4 = base + i * THREADS;
        if (i4 < N4) {
            __builtin_amdgcn_global_load_async_to_lds_b128(
                (gbl_v4i_p)&xb[i4],
                (lds_v4i_p)&stage[i * THREADS + threadIdx.x],
                0, 0);
        }
    }
#endif

#pragma unroll
    for (int i = 0; i < ITEMS; ++i) {
        const int i4 = base + i * THREADS;
        if (i4 < N4) {
            const int t  = i4 / ROW4;            // time row of this float4
            const int f0 = (i4 - t * ROW4) * 4;  // first freq bin (16B aligned)
            const bool tm = (t >= ts0 && t < te0) || (t >= ts1 && t < te1);

#if USE_ASYNC
            // Consume in issue order; argument must be a literal constant, so
            // dispatch on (compile-time-unrolled) i. Younger loads stay in flight.
            if      (i == 0) __builtin_amdgcn_s_wait_asynccnt(ITEMS - 1);
            else if (i == 1) __builtin_amdgcn_s_wait_asynccnt(ITEMS - 2);
            else if (i == 2) __builtin_amdgcn_s_wait_asynccnt(ITEMS - 3);
            else             __builtin_amdgcn_s_wait_asynccnt(0);
            v4f v = stage[i * THREADS + threadIdx.x];
#else
            v4f v = xb[i4];
#endif
            v4f k = *(const v4f*)&kf[f0];        // ds_load_b128
            const float kt = tm ? 0.0f : 1.0f;   // matches reference x*(kf*kt)
            v4f r = v * k * kt;
            // NT store: streamed-once output must not evict the (L2-resident,
            // 163.8MB < 192MB) input between graph replays.
            __builtin_nontemporal_store(r, &ob[i4]);
        }
    }
}

extern "C" void kernel_launch(void* const* d_in, const int* in_sizes, int n_in,
                              void* d_out, int out_size, void* d_ws, size_t ws_size,
                              hipStream_t stream) {
    (void)n_in; (void)out_size; (void)d_ws; (void)ws_size;
    const float* x  = (const float*)d_in[0];
    const int* fw   = (const int*)d_in[1];
    const int* fs   = (const int*)d_in[2];
    const int* tw   = (const int*)d_in[3];
    const int* ts   = (const int*)d_in[4];
    float* out      = (float*)d_out;

    const int bdim = in_sizes[1] / 2;  // 128 per reference shapes
    dim3 grid((N4 + TILE - 1) / TILE, bdim);
    SpecAugment_75239237092009_kernel<<<grid, THREADS, 0, stream>>>(
        x, fw, fs, tw, ts, out);
}